// ConstraintGraphEncoder_75780402970977
// MI455X (gfx1250) — compile-verified
//
#include <hip/hip_runtime.h>
#include <hip/hip_bf16.h>
#include <math.h>

typedef float v2f __attribute__((ext_vector_type(2)));
typedef float v8f __attribute__((ext_vector_type(8)));
typedef unsigned int u32x4 __attribute__((ext_vector_type(4)));
typedef int i32x4 __attribute__((ext_vector_type(4)));
typedef int i32x8 __attribute__((ext_vector_type(8)));

#define NEG_SLOPE 0.2f
#define KC 64          // K-chunk staged in LDS
#define BSTRIDE 65     // LDS row stride (pad to avoid half-wave bank conflicts)

#if __has_builtin(__builtin_amdgcn_tensor_load_to_lds)
#define HAVE_TDM 1
#else
#define HAVE_TDM 0
#endif

// ---- monotone float<->uint key (for atomicMax-based segment max, handles negatives) ----
static __device__ __forceinline__ unsigned fkey(float f) {
  unsigned u = __float_as_uint(f);
  return (u & 0x80000000u) ? ~u : (u | 0x80000000u);
}
static __device__ __forceinline__ float fkeyinv(unsigned k) {
  unsigned u = (k & 0x80000000u) ? (k ^ 0x80000000u) : ~k;
  return __uint_as_float(u);
}

// =======================================================================
// TDM: async-load a [kchunk x 64] f32 tile (row stride Nc elements) from
// global into LDS at byte offset ldsOff, padding 1 DWORD per 64 DWORDs so
// the LDS layout has row stride BSTRIDE=65 floats. D# per CDNA5 ISA §8.3/8.4.
// Issued once per executing wave (EXEC ignored); tracked by TENSORcnt.
// =======================================================================
#if HAVE_TDM
static __device__ __forceinline__ void tdm_load_b_tile(const float* gsrc,
                                                       unsigned ldsOff,
                                                       int Nc, int kchunk)
{
  const unsigned long long ga = (unsigned long long)(size_t)gsrc;
  u32x4 g0;
  g0[0] = 1u;                                        // count=1, user mode, no gather
  g0[1] = ldsOff;                                    // lds_addr (bytes)
  g0[2] = (unsigned)(ga & 0xFFFFFFFFull);            // global_addr[31:0]
  g0[3] = (unsigned)((ga >> 32) & 0x1FFFFFFull)      // global_addr[56:32]
        | (2u << 30);                                // type = 2 ("image")
  unsigned w0 = (2u << 16)                           // data_size = 4 bytes
              | (1u << 20)                           // pad_enable
              | (5u << 22)                           // pad_interval: 64 DWORDs
              | (0u << 25);                          // pad_amount: 1 DWORD
  unsigned td0 = (unsigned)Nc;                       // tensor_dim0
  unsigned td1 = (unsigned)kchunk;                   // tensor_dim1
  unsigned st0 = (unsigned)Nc;                       // tensor_dim0_stride
  i32x8 g1;
  g1[0] = (int)w0;
  g1[1] = (int)((td0 & 0xFFFFu) << 16);              // barrier_addr=0 | tensor_dim0[15:0]
  g1[2] = (int)(((td0 >> 16) & 0xFFFFu) | ((td1 & 0xFFFFu) << 16));
  g1[3] = (int)(((td1 >> 16) & 0xFFFFu) | (64u << 16));   // tile_dim0 = 64
  g1[4] = (int)(td1 & 0xFFFFu);                      // tile_dim1 = kchunk, tile_dim2 = 0
  g1[5] = (int)st0;                                  // tensor_dim0_stride[31:0]
  g1[6] = 0;                                         // stride[47:32] | dim1_stride[15:0]
  g1[7] = 0;
  i32x4 g2 = {0, 0, 0, 0};
  i32x4 g3 = {0, 0, 0, 0};
#if defined(__clang_major__) && (__clang_major__ >= 23)
  i32x8 g4 = {0, 0, 0, 0, 0, 0, 0, 0};
  __builtin_amdgcn_tensor_load_to_lds(g0, g1, g2, g3, g4, 0);
#else
  __builtin_amdgcn_tensor_load_to_lds(g0, g1, g2, g3, 0);
#endif
}
#endif

// =======================================================================
// WMMA f32 GEMM:  C[MxNc] = act( A[MxK] @ B[KxNc] + bias ),  act: 0 none, 1 relu
// Block = 256 threads = 8 waves -> 128x64 output tile; each wave a 16x64 strip.
// TDM variant: double-buffered async B staging (tensor_load_to_lds +
// s_wait_tensorcnt) overlapped with WMMA on the other buffer.
// Requires: Nc % 64 == 0, K % KC == 0. Grid: (ceil(M/128), Nc/64).
// =======================================================================
__global__ void gemm_f32_wmma_tdm(const float* __restrict__ A, const float* __restrict__ B,
                                  const float* __restrict__ bias, float* __restrict__ Cmat,
                                  int M, int K, int Nc, int act)
{
  __shared__ float Bs[2 * KC * BSTRIDE];
  const int tid  = threadIdx.x;
  const int wave = tid >> 5;
  const int lane = tid & 31;
  const int half = lane >> 4;
  const int l16  = lane & 15;

  const int row0  = blockIdx.x * 128 + wave * 16;
  const int ncol0 = blockIdx.y * 64;
  const int arow  = row0 + l16;
  const int arowc = (arow < M) ? arow : (M - 1);

  v8f acc[4];
  #pragma unroll
  for (int j = 0; j < 4; ++j) acc[j] = (v8f){0.f,0.f,0.f,0.f,0.f,0.f,0.f,0.f};

  const int nch = K / KC;

#if HAVE_TDM
  const unsigned ldsBase = (unsigned)((size_t)(void*)&Bs[0] & 0xFFFFFFFFull);
  if (wave == 0)
    tdm_load_b_tile(B + ncol0, ldsBase, Nc, KC);
  for (int c = 0; c < nch; ++c) {
    if (wave == 0) {
      if (c + 1 < nch) {
        tdm_load_b_tile(B + (size_t)(c + 1) * KC * Nc + ncol0,
                        ldsBase + (unsigned)(((c + 1) & 1) * (KC * BSTRIDE * 4)),
                        Nc, KC);
        __builtin_amdgcn_s_wait_tensorcnt(1);   // chunk c complete (in-order TDM)
      } else {
        __builtin_amdgcn_s_wait_tensorcnt(0);   // last chunk complete
      }
    }
    __syncthreads();                       // chunk c visible to all waves
    const int boff = (c & 1) * (KC * BSTRIDE);
    const float* Arow = A + (size_t)arowc * K + (size_t)c * KC;
    __builtin_prefetch(Arow + KC, 0, 3);   // global_prefetch_b8: next A chunk
    #pragma unroll 4
    for (int kk0 = 0; kk0 < KC; kk0 += 4) {
      const int ka = kk0 + (half << 1);
      v2f a = *reinterpret_cast<const v2f*>(Arow + ka);
      #pragma unroll
      for (int j = 0; j < 4; ++j) {
        v2f b;
        b.x = Bs[boff + ka * BSTRIDE + (j << 4) + l16];
        b.y = Bs[boff + (ka + 1) * BSTRIDE + (j << 4) + l16];
        acc[j] = __builtin_amdgcn_wmma_f32_16x16x4_f32(false, a, false, b, (short)0,
                                                       acc[j], false, false);
      }
    }
    __syncthreads();                       // all reads of this buffer done
  }
#else
  // cooperative-staging fallback (no TDM builtin on this toolchain)
  for (int c = 0; c < nch; ++c) {
    __syncthreads();
    for (int t = tid; t < KC * 64; t += 256) {
      const int kk = t >> 6, cc = t & 63;
      Bs[kk * BSTRIDE + cc] = B[(size_t)(c * KC + kk) * Nc + ncol0 + cc];
    }
    __syncthreads();
    const float* Arow = A + (size_t)arowc * K + (size_t)c * KC;
    __builtin_prefetch(Arow + KC, 0, 3);
    #pragma unroll 4
    for (int kk0 = 0; kk0 < KC; kk0 += 4) {
      const int ka = kk0 + (half << 1);
      v2f a = *reinterpret_cast<const v2f*>(Arow + ka);
      #pragma unroll
      for (int j = 0; j < 4; ++j) {
        v2f b;
        b.x = Bs[ka * BSTRIDE + (j << 4) + l16];
        b.y = Bs[(ka + 1) * BSTRIDE + (j << 4) + l16];
        acc[j] = __builtin_amdgcn_wmma_f32_16x16x4_f32(false, a, false, b, (short)0,
                                                       acc[j], false, false);
      }
    }
  }
#endif

  #pragma unroll
  for (int j = 0; j < 4; ++j) {
    const int col = ncol0 + (j << 4) + l16;
    const float bv = bias ? bias[col] : 0.0f;
    #pragma unroll
    for (int v = 0; v < 8; ++v) {
      const int row = row0 + v + (half << 3);
      if (row < M) {
        float xv = acc[j][v] + bv;
        if (act == 1) xv = xv > 0.0f ? xv : 0.0f;
        Cmat[(size_t)row * Nc + col] = xv;
      }
    }
  }
}

// -------- manual-staging GEMM (handles K % KC != 0; same tiling) --------
__global__ void gemm_f32_wmma(const float* __restrict__ A, const float* __restrict__ B,
                              const float* __restrict__ bias, float* __restrict__ Cmat,
                              int M, int K, int Nc, int act)
{
  __shared__ float Bs[KC * BSTRIDE];
  const int tid  = threadIdx.x;
  const int wave = tid >> 5;
  const int lane = tid & 31;
  const int half = lane >> 4;
  const int l16  = lane & 15;

  const int row0  = blockIdx.x * 128 + wave * 16;
  const int ncol0 = blockIdx.y * 64;
  const int arow  = row0 + l16;
  const int arowc = (arow < M) ? arow : (M - 1);

  v8f acc[4];
  #pragma unroll
  for (int j = 0; j < 4; ++j) acc[j] = (v8f){0.f,0.f,0.f,0.f,0.f,0.f,0.f,0.f};

  for (int kc0 = 0; kc0 < K; kc0 += KC) {
    const int kchunk = (K - kc0) < KC ? (K - kc0) : KC;
    __syncthreads();
    for (int t = tid; t < kchunk * 64; t += 256) {
      const int kk = t >> 6, cc = t & 63;
      Bs[kk * BSTRIDE + cc] = B[(size_t)(kc0 + kk) * Nc + ncol0 + cc];
    }
    __syncthreads();
    const float* Arow = A + (size_t)arowc * K + kc0;
    __builtin_prefetch(Arow + KC, 0, 3);
    #pragma unroll 4
    for (int kk0 = 0; kk0 < kchunk; kk0 += 4) {
      const int ka = kk0 + (half << 1);
      v2f a = *reinterpret_cast<const v2f*>(Arow + ka);
      #pragma unroll
      for (int j = 0; j < 4; ++j) {
        v2f b;
        b.x = Bs[ka * BSTRIDE + (j << 4) + l16];
        b.y = Bs[(ka + 1) * BSTRIDE + (j << 4) + l16];
        acc[j] = __builtin_amdgcn_wmma_f32_16x16x4_f32(false, a, false, b, (short)0,
                                                       acc[j], false, false);
      }
    }
  }

  #pragma unroll
  for (int j = 0; j < 4; ++j) {
    const int col = ncol0 + (j << 4) + l16;
    const float bv = bias ? bias[col] : 0.0f;
    #pragma unroll
    for (int v = 0; v < 8; ++v) {
      const int row = row0 + v + (half << 3);
      if (row < M) {
        float xv = acc[j][v] + bv;
        if (act == 1) xv = xv > 0.0f ? xv : 0.0f;
        Cmat[(size_t)row * Nc + col] = xv;
      }
    }
  }
}

// =======================================================================
// Attention logits: asrc[n,h] = sum_c feat[n,h,c]*avs[h,c]; same for adst.
// =======================================================================
__global__ void alpha_kernel(const float* __restrict__ feat,
                             const float* __restrict__ avs, const float* __restrict__ avd,
                             float* __restrict__ asrc, float* __restrict__ adst,
                             int NH, int H, int C)
{
  const int wave = (int)((blockIdx.x * blockDim.x + threadIdx.x) >> 5);
  const int lane = threadIdx.x & 31;
  if (wave >= NH) return;                // wave-uniform
  const int h = wave % H;
  const float* f = feat + (size_t)wave * C;
  float s1 = 0.f, s2 = 0.f;
  for (int c = lane; c < C; c += 32) {
    const float v = f[c];
    s1 += v * avs[h * C + c];
    s2 += v * avd[h * C + c];
  }
  #pragma unroll
  for (int o = 16; o > 0; o >>= 1) {
    s1 += __shfl_xor(s1, o, 32);
    s2 += __shfl_xor(s2, o, 32);
  }
  if (lane == 0) { asrc[wave] = s1; adst[wave] = s2; }
}

// =======================================================================
// Edge passes: e < E come from edge_index; e >= E are self-loops.
// =======================================================================
static __device__ __forceinline__ void edge_nodes(const int* __restrict__ ei,
                                                  size_t e, int E, int& s, int& d)
{
  if (e < (size_t)E) { s = ei[e]; d = ei[(size_t)E + e]; }
  else { s = (int)(e - (size_t)E); d = s; }
}

template<int H>
__global__ void edge_max_t(const int* __restrict__ ei, int E, int N,
                           const float* __restrict__ asrc, const float* __restrict__ adst,
                           unsigned* __restrict__ mkey)
{
  const size_t t = blockIdx.x * (size_t)blockDim.x + threadIdx.x;
  const size_t total = (size_t)(E + N) * H;
  if (t >= total) return;
  const size_t e = t / H;
  const int h = (int)(t - e * H);
  int s, d; edge_nodes(ei, e, E, s, d);
  float x = asrc[(size_t)s * H + h] + adst[(size_t)d * H + h];
  x = x > 0.0f ? x : NEG_SLOPE * x;
  atomicMax(&mkey[(size_t)d * H + h], fkey(x));
}

template<int H>
__global__ void edge_exp_t(const int* __restrict__ ei, int E, int N,
                           const float* __restrict__ asrc, const float* __restrict__ adst,
                           const unsigned* __restrict__ mkey,
                           float* __restrict__ ealpha, float* __restrict__ ssum)
{
  const size_t t = blockIdx.x * (size_t)blockDim.x + threadIdx.x;
  const size_t total = (size_t)(E + N) * H;
  if (t >= total) return;
  const size_t e = t / H;
  const int h = (int)(t - e * H);
  int s, d; edge_nodes(ei, e, E, s, d);
  float x = asrc[(size_t)s * H + h] + adst[(size_t)d * H + h];
  x = x > 0.0f ? x : NEG_SLOPE * x;
  const float m = fkeyinv(mkey[(size_t)d * H + h]);
  const float ex = __expf(x - m);
  ealpha[t] = ex;
  atomicAdd(&ssum[(size_t)d * H + h], ex);
}

template<int H>
__global__ void edge_norm_t(const int* __restrict__ ei, int E, int N,
                            const float* __restrict__ ssum, float* __restrict__ ealpha)
{
  const size_t t = blockIdx.x * (size_t)blockDim.x + threadIdx.x;
  const size_t total = (size_t)(E + N) * H;
  if (t >= total) return;
  const size_t e = t / H;
  const int h = (int)(t - e * H);
  int s, d; edge_nodes(ei, e, E, s, d);
  (void)s;
  ealpha[t] = ealpha[t] / (ssum[(size_t)d * H + h] + 1e-16f);
}

template<int H, int C>
__global__ void aggregate_t(const int* __restrict__ ei, int E, int N,
                            const float* __restrict__ feat,
                            const float* __restrict__ ealpha,
                            float* __restrict__ agg)
{
  const size_t t = blockIdx.x * (size_t)blockDim.x + threadIdx.x;
  const size_t HC = (size_t)H * C;
  const size_t total = (size_t)(E + N) * HC;
  if (t >= total) return;
  const size_t e = t / HC;                 // power-of-two constant -> shift
  const int r = (int)(t - e * HC);
  const int h = r / C;
  int s, d; edge_nodes(ei, e, E, s, d);
  const float a = ealpha[e * H + h];
  atomicAdd(&agg[(size_t)d * HC + r], feat[(size_t)s * HC + r] * a);
}

// -------- generic fallbacks (runtime H, C) --------
__global__ void edge_max_g(const int* __restrict__ ei, int E, int N, int H,
                           const float* __restrict__ asrc, const float* __restrict__ adst,
                           unsigned* __restrict__ mkey)
{
  const size_t t = blockIdx.x * (size_t)blockDim.x + threadIdx.x;
  const size_t total = (size_t)(E + N) * H;
  if (t >= total) return;
  const size_t e = t / H; const int h = (int)(t - e * H);
  int s, d; edge_nodes(ei, e, E, s, d);
  float x = asrc[(size_t)s * H + h] + adst[(size_t)d * H + h];
  x = x > 0.0f ? x : NEG_SLOPE * x;
  atomicMax(&mkey[(size_t)d * H + h], fkey(x));
}
__global__ void edge_exp_g(const int* __restrict__ ei, int E, int N, int H,
                           const float* __restrict__ asrc, const float* __restrict__ adst,
                           const unsigned* __restrict__ mkey,
                           float* __restrict__ ealpha, float* __restrict__ ssum)
{
  const size_t t = blockIdx.x * (size_t)blockDim.x + threadIdx.x;
  const size_t total = (size_t)(E + N) * H;
  if (t >= total) return;
  const size_t e = t / H; const int h = (int)(t - e * H);
  int s, d; edge_nodes(ei, e, E, s, d);
  float x = asrc[(size_t)s * H + h] + adst[(size_t)d * H + h];
  x = x > 0.0f ? x : NEG_SLOPE * x;
  const float m = fkeyinv(mkey[(size_t)d * H + h]);
  const float ex = __expf(x - m);
  ealpha[t] = ex;
  atomicAdd(&ssum[(size_t)d * H + h], ex);
}
__global__ void edge_norm_g(const int* __restrict__ ei, int E, int N, int H,
                            const float* __restrict__ ssum, float* __restrict__ ealpha)
{
  const size_t t = blockIdx.x * (size_t)blockDim.x + threadIdx.x;
  const size_t total = (size_t)(E + N) * H;
  if (t >= total) return;
  const size_t e = t / H; const int h = (int)(t - e * H);
  int s, d; edge_nodes(ei, e, E, s, d);
  (void)s;
  ealpha[t] = ealpha[t] / (ssum[(size_t)d * H + h] + 1e-16f);
}
__global__ void aggregate_g(const int* __restrict__ ei, int E, int N, int H, int C,
                            const float* __restrict__ feat,
                            const float* __restrict__ ealpha, float* __restrict__ agg)
{
  const size_t t = blockIdx.x * (size_t)blockDim.x + threadIdx.x;
  const size_t HC = (size_t)H * C;
  const size_t total = (size_t)(E + N) * HC;
  if (t >= total) return;
  const size_t e = t / HC; const int r = (int)(t - e * HC); const int h = r / C;
  int s, d; edge_nodes(ei, e, E, s, d);
  atomicAdd(&agg[(size_t)d * HC + r], feat[(size_t)s * HC + r] * ealpha[e * H + h]);
}

// x = elu(x + bias[i % D])
__global__ void bias_elu_kernel(float* __restrict__ x, const float* __restrict__ bias,
                                size_t total, int D)
{
  const size_t t = blockIdx.x * (size_t)blockDim.x + threadIdx.x;
  if (t >= total) return;
  float v = x[t] + bias[t % (size_t)D];
  x[t] = v > 0.0f ? v : expm1f(v);
}

// column sums for global mean pool
__global__ void colsum_kernel(const float* __restrict__ x, float* __restrict__ g,
                              int N, int C)
{
  const int c = threadIdx.x % C;
  const int rg = threadIdx.x / C;
  const int rpb = blockDim.x / C;
  float acc = 0.f;
  for (int r = blockIdx.x * rpb + rg; r < N; r += gridDim.x * rpb)
    acc += x[(size_t)r * C + c];
  atomicAdd(&g[c], acc);
}

// out[j] = b_out[j] + sum_c (g[c] * invN) * Wout[c, j]
__global__ void final_out_kernel(const float* __restrict__ g, const float* __restrict__ Wout,
                                 const float* __restrict__ bout, float* __restrict__ out,
                                 int C, int Dout, float invN)
{
  const int j = blockIdx.x * blockDim.x + threadIdx.x;
  if (j >= Dout) return;
  float acc = bout[j];
  for (int c = 0; c < C; ++c) acc += g[c] * invN * Wout[(size_t)c * Dout + j];
  out[j] = acc;
}

// =======================================================================
// Host orchestration
// =======================================================================
static inline unsigned cdiv(size_t a, size_t b) { return (unsigned)((a + b - 1) / b); }

static inline void launch_gemm(const float* A, const float* B, const float* bias,
                               float* Cmat, int M, int K, int Nc, int act,
                               hipStream_t stream)
{
  dim3 grid(cdiv(M, 128), Nc / 64);
  if ((K % KC) == 0)
    gemm_f32_wmma_tdm<<<grid, 256, 0, stream>>>(A, B, bias, Cmat, M, K, Nc, act);
  else
    gemm_f32_wmma<<<grid, 256, 0, stream>>>(A, B, bias, Cmat, M, K, Nc, act);
}

extern "C" void kernel_launch(void* const* d_in, const int* in_sizes, int n_in,
                              void* d_out, int out_size, void* d_ws, size_t ws_size,
                              hipStream_t stream)
{
  const float* x      = (const float*)d_in[0];
  const int*   ei     = (const int*)  d_in[1];
  const float* W_emb  = (const float*)d_in[2];
  const float* b_emb  = (const float*)d_in[3];
  const float* W1     = (const float*)d_in[4];
  const float* a1_src = (const float*)d_in[5];
  const float* a1_dst = (const float*)d_in[6];
  const float* b1     = (const float*)d_in[7];
  const float* W2     = (const float*)d_in[8];
  const float* a2_src = (const float*)d_in[9];
  const float* a2_dst = (const float*)d_in[10];
  const float* b2     = (const float*)d_in[11];
  const float* W_out  = (const float*)d_in[12];
  const float* b_out  = (const float*)d_in[13];
  float* out = (float*)d_out;

  const int hidden   = in_sizes[3];                     // 128
  const int node_dim = in_sizes[2] / hidden;            // 64
  const int N        = in_sizes[0] / node_dim;          // 25000
  const int E        = in_sizes[1] / 2;                 // 400000
  const int heads    = in_sizes[4] / (hidden * hidden); // 4
  const int out_dim  = in_sizes[13];                    // 256
  const int HC1      = heads * hidden;                  // 512
  const size_t Etot  = (size_t)E + N;
  const bool fast    = (hidden == 128);

  // ---- workspace layout (floats) ----
  float* W = (float*)d_ws;
  size_t o = 0;
  float* h0   = W + o; o += (size_t)N * hidden;   // [N,128]
  float* h1   = W + o; o += (size_t)N * HC1;      // [N,512]
  float* agg1 = W + o; o += (size_t)N * HC1;      // [N,512] -> elu'd in place
  float* agg2 = W + o; o += (size_t)N * hidden;   // [N,128]
  float* asrc1 = W + o; o += (size_t)N * heads;
  float* adst1 = W + o; o += (size_t)N * heads;
  unsigned* mkey1 = (unsigned*)(W + o); o += (size_t)N * heads;
  float* ssum1 = W + o; o += (size_t)N * heads;
  float* asrc2 = W + o; o += (size_t)N;
  float* adst2 = W + o; o += (size_t)N;
  unsigned* mkey2 = (unsigned*)(W + o); o += (size_t)N;
  float* ssum2 = W + o; o += (size_t)N;
  float* ealpha = W + o; o += Etot * heads;       // reused for layer 2
  float* g = W + o; o += (size_t)hidden;
  float* h2 = h0;                                  // reuse: h0 dead after h1 GEMM

  const int BLK = 256;

  // ===================== Embedding: h0 = relu(x @ W_emb + b_emb) =====================
  launch_gemm(x, W_emb, b_emb, h0, N, node_dim, hidden, 1, stream);

  // ===================== GAT layer 1 (heads=4, concat) =====================
  launch_gemm(h0, W1, nullptr, h1, N, hidden, HC1, 0, stream);
  alpha_kernel<<<cdiv((size_t)N * heads, 8), BLK, 0, stream>>>(h1, a1_src, a1_dst,
                                                               asrc1, adst1,
                                                               N * heads, heads, hidden);
  (void)hipMemsetAsync(mkey1, 0, (size_t)N * heads * 4, stream);
  (void)hipMemsetAsync(ssum1, 0, (size_t)N * heads * 4, stream);
  (void)hipMemsetAsync(agg1,  0, (size_t)N * HC1 * 4, stream);
  {
    const size_t eh = Etot * heads;
    const size_t tw = Etot * HC1;
    if (heads == 4 && fast) {
      edge_max_t<4> <<<cdiv(eh, BLK), BLK, 0, stream>>>(ei, E, N, asrc1, adst1, mkey1);
      edge_exp_t<4> <<<cdiv(eh, BLK), BLK, 0, stream>>>(ei, E, N, asrc1, adst1,
                                                        mkey1, ealpha, ssum1);
      edge_norm_t<4><<<cdiv(eh, BLK), BLK, 0, stream>>>(ei, E, N, ssum1, ealpha);
      aggregate_t<4,128><<<cdiv(tw, BLK), BLK, 0, stream>>>(ei, E, N, h1, ealpha, agg1);
    } else {
      edge_max_g <<<cdiv(eh, BLK), BLK, 0, stream>>>(ei, E, N, heads, asrc1, adst1, mkey1);
      edge_exp_g <<<cdiv(eh, BLK), BLK, 0, stream>>>(ei, E, N, heads, asrc1, adst1,
                                                     mkey1, ealpha, ssum1);
      edge_norm_g<<<cdiv(eh, BLK), BLK, 0, stream>>>(ei, E, N, heads, ssum1, ealpha);
      aggregate_g<<<cdiv(tw, BLK), BLK, 0, stream>>>(ei, E, N, heads, hidden,
                                                     h1, ealpha, agg1);
    }
  }
  bias_elu_kernel<<<cdiv((size_t)N * HC1, BLK), BLK, 0, stream>>>(agg1, b1,
                                                                  (size_t)N * HC1, HC1);

  // ===================== GAT layer 2 (heads=1, mean == identity) =====================
  launch_gemm(agg1, W2, nullptr, h2, N, HC1, hidden, 0, stream);
  alpha_kernel<<<cdiv((size_t)N, 8), BLK, 0, stream>>>(h2, a2_src, a2_dst,
                                                       asrc2, adst2, N, 1, hidden);
  (void)hipMemsetAsync(mkey2, 0, (size_t)N * 4, stream);
  (void)hipMemsetAsync(ssum2, 0, (size_t)N * 4, stream);
  (void)hipMemsetAsync(agg2,  0, (size_t)N * hidden * 4, stream);
  (void)hipMemsetAsync(g,     0, (size_t)hidden * 4, stream);
  {
    const size_t tw = Etot * hidden;
    if (fast) {
      edge_max_t<1> <<<cdiv(Etot, BLK), BLK, 0, stream>>>(ei, E, N, asrc2, adst2, mkey2);
      edge_exp_t<1> <<<cdiv(Etot, BLK), BLK, 0, stream>>>(ei, E, N, asrc2, adst2,
                                                          mkey2, ealpha, ssum2);
      edge_norm_t<1><<<cdiv(Etot, BLK), BLK, 0, stream>>>(ei, E, N, ssum2, ealpha);
      aggregate_t<1,128><<<cdiv(tw, BLK), BLK, 0, stream>>>(ei, E, N, h2, ealpha, agg2);
    } else {
      edge_max_g <<<cdiv(Etot, BLK), BLK, 0, stream>>>(ei, E, N, 1, asrc2, adst2, mkey2);
      edge_exp_g <<<cdiv(Etot, BLK), BLK, 0, stream>>>(ei, E, N, 1, asrc2, adst2,
                                                       mkey2, ealpha, ssum2);
      edge_norm_g<<<cdiv(Etot, BLK), BLK, 0, stream>>>(ei, E, N, 1, ssum2, ealpha);
      aggregate_g<<<cdiv(tw, BLK), BLK, 0, stream>>>(ei, E, N, 1, hidden,
                                                     h2, ealpha, agg2);
    }
  }
  bias_elu_kernel<<<cdiv((size_t)N * hidden, BLK), BLK, 0, stream>>>(agg2, b2,
                                                                     (size_t)N * hidden,
                                                                     hidden);

  // ===================== Mean pool + output projection =====================
  colsum_kernel<<<512, BLK, 0, stream>>>(agg2, g, N, hidden);
  final_out_kernel<<<cdiv(out_dim, BLK), BLK, 0, stream>>>(g, W_out, b_out, out,
                                                           hidden, out_dim, 1.0f / N);
}